// CosineLobeSampler_60679297958361
// MI455X (gfx1250) — compile-verified
//
#include <hip/hip_runtime.h>
#include <hip/hip_bf16.h>

#define PI_F 3.14159265358979323846f

// All three normalizations in this kernel have norms bounded well away from
// zero (tangent >= 0.436, bitangent ~1, |H_l| >= 1 - r1 >= 0.05), so the
// reference's x/(||x||+1e-8) is within ~2 ulp of x * rsqrt(||x||^2).
// Use the native v_rsq_f32 (~1 ulp) instead of v_sqrt + full IEEE divide
// (~18 instrs) to keep instruction issue under the HBM-bandwidth floor.
__device__ __forceinline__ float fast_rnorm3(float x, float y, float z)
{
    return __builtin_amdgcn_rsqf(x * x + y * y + z * z);
}

__device__ __forceinline__ void compute_basis(float nx, float ny, float nz,
                                              float& tx, float& ty, float& tz,
                                              float& qx, float& qy, float& qz)
{
    // up = (normal.z < 0.9) ? (0,0,1) : (-1,0,0)
    const bool zup = (nz < 0.9f);
    const float upx = zup ? 0.0f : -1.0f;
    const float upz = zup ? 1.0f : 0.0f;
    // tangent = normalize(cross(up, n));  up.y == 0 always
    tx =          - upz * ny;
    ty = upz * nx - upx * nz;
    tz = upx * ny;
    const float ti = fast_rnorm3(tx, ty, tz);
    tx *= ti; ty *= ti; tz *= ti;
    // bitangent = normalize(cross(n, tangent))
    qx = ny * tz - nz * ty;
    qy = nz * tx - nx * tz;
    qz = nx * ty - ny * tx;
    const float qi = fast_rnorm3(qx, qy, qz);
    qx *= qi; qy *= qi; qz *= qi;
}

__device__ __forceinline__ void sample_and_store(
    float u1, float u2, float m, bool first_sample,
    float rr, float nx, float ny, float nz,
    float vx, float vy, float vz,
    float tx, float ty, float tz,
    float qx, float qy, float qz,
    float* __restrict__ outL, long lbase,
    float* __restrict__ outB, long bbase)
{
    // sphere_noise = (ct*cp, ct*sp, -st);  H_l = normalize(r1*noise + e3)
    const float theta = u1 * PI_F;
    const float phi   = u2 * (2.0f * PI_F);
    const float st = __sinf(theta), ct = __cosf(theta);
    const float sp = __sinf(phi),   cp = __cosf(phi);
    float hx = rr * ct * cp;
    float hy = rr * ct * sp;
    float hz = 1.0f - rr * st;
    const float hi = fast_rnorm3(hx, hy, hz);
    hx *= hi; hy *= hi; hz *= hi;
    if (first_sample) { hx = 0.0f; hy = 0.0f; hz = 1.0f; }   // H_l[:,0,:] = e3
    // H[i] = H_l.x*tangent[i] + H_l.y*bitangent[i] + H_l.z*normal[i]
    const float Hx = hx * tx + hy * qx + hz * nx;
    const float Hy = hx * ty + hy * qy + hz * ny;
    const float Hz = hx * tz + hy * qz + hz * nz;
    const float d2 = 2.0f * (vx * Hx + vy * Hy + vz * Hz);
    // L = (2(V.H)H - V) * m   — streaming outputs: non-temporal stores
    __builtin_nontemporal_store((d2 * Hx - vx) * m, &outL[lbase + 0]);
    __builtin_nontemporal_store((d2 * Hy - vy) * m, &outL[lbase + 1]);
    __builtin_nontemporal_store((d2 * Hz - vz) * m, &outL[lbase + 2]);
    // basis_out[i][j] = basis[j][i] * m  (i-major)
    __builtin_nontemporal_store(tx * m, &outB[bbase + 0]);
    __builtin_nontemporal_store(qx * m, &outB[bbase + 1]);
    __builtin_nontemporal_store(nx * m, &outB[bbase + 2]);
    __builtin_nontemporal_store(ty * m, &outB[bbase + 3]);
    __builtin_nontemporal_store(qy * m, &outB[bbase + 4]);
    __builtin_nontemporal_store(ny * m, &outB[bbase + 5]);
    __builtin_nontemporal_store(tz * m, &outB[bbase + 6]);
    __builtin_nontemporal_store(qz * m, &outB[bbase + 7]);
    __builtin_nontemporal_store(nz * m, &outB[bbase + 8]);
}

// Specialized S == 64: wave-uniform row index -> scalar-path row loads;
// async-to-LDS staging of the streamed per-sample inputs (CDNA5 ASYNCcnt).
__global__ __launch_bounds__(256) void lobe_s64_kernel(
    const float* __restrict__ viewdir,
    const float* __restrict__ normal,
    const float* __restrict__ r1,
    const int*   __restrict__ ray_mask,
    const float* __restrict__ angs,
    float* __restrict__ outL,
    float* __restrict__ outB)
{
    __shared__ float s_ang[256 * 2];
    __shared__ int   s_msk[256];

    const int  tid = threadIdx.x;
    const long gid = (long)blockIdx.x * 256 + tid;   // total is a multiple of 256

#if defined(__gfx1250__)
    {
        // Kick off async copies (global -> LDS, per-lane addresses) before the
        // basis math so the streamed traffic overlaps the uniform VALU/SALU work.
        unsigned lds_a = (unsigned)(size_t)&s_ang[tid * 2];
        unsigned lds_m = (unsigned)(size_t)&s_msk[tid];
        unsigned off_a = (unsigned)(gid * 8);   // byte offset into angs (float2/lane)
        unsigned off_m = (unsigned)(gid * 4);   // byte offset into ray_mask (int/lane)
        asm volatile("global_load_async_to_lds_b64 %0, %1, %2"
                     :: "v"(lds_a), "v"(off_a), "s"(angs) : "memory");
        asm volatile("global_load_async_to_lds_b32 %0, %1, %2"
                     :: "v"(lds_m), "v"(off_m), "s"(ray_mask) : "memory");
    }
#endif

    // Row index is uniform across each wave32 (64 consecutive threads / row):
    // force the per-row fetches onto the scalar (s_load / KMcnt) path.
    const int b = __builtin_amdgcn_readfirstlane((int)(gid >> 6));

    const float nx = normal[3 * b + 0];
    const float ny = normal[3 * b + 1];
    const float nz = normal[3 * b + 2];
    const float vx = viewdir[3 * b + 0];
    const float vy = viewdir[3 * b + 1];
    const float vz = viewdir[3 * b + 2];
    const float rr = r1[b];

    float tx, ty, tz, qx, qy, qz;
    compute_basis(nx, ny, nz, tx, ty, tz, qx, qy, qz);

    float u1, u2, m;
#if defined(__gfx1250__)
    asm volatile("s_wait_asynccnt 0" ::: "memory");
    u1 = s_ang[tid * 2 + 0];
    u2 = s_ang[tid * 2 + 1];
    m  = s_msk[tid] ? 1.0f : 0.0f;
#else
    u1 = angs[gid * 2 + 0];
    u2 = angs[gid * 2 + 1];
    m  = ray_mask[gid] ? 1.0f : 0.0f;
#endif

    sample_and_store(u1, u2, m, (gid & 63) == 0,
                     rr, nx, ny, nz, vx, vy, vz,
                     tx, ty, tz, qx, qy, qz,
                     outL, gid * 3, outB, gid * 9);
}

// Generic-S fallback (plain loads, bounds check).
__global__ __launch_bounds__(256) void lobe_generic_kernel(
    const float* __restrict__ viewdir,
    const float* __restrict__ normal,
    const float* __restrict__ r1,
    const int*   __restrict__ ray_mask,
    const float* __restrict__ angs,
    float* __restrict__ outL,
    float* __restrict__ outB,
    int S, long total)
{
    const long gid = (long)blockIdx.x * blockDim.x + threadIdx.x;
    if (gid >= total) return;
    const int b = (int)(gid / S);
    const int s = (int)(gid - (long)b * S);

    const float nx = normal[3 * b + 0], ny = normal[3 * b + 1], nz = normal[3 * b + 2];
    const float vx = viewdir[3 * b + 0], vy = viewdir[3 * b + 1], vz = viewdir[3 * b + 2];
    const float rr = r1[b];

    float tx, ty, tz, qx, qy, qz;
    compute_basis(nx, ny, nz, tx, ty, tz, qx, qy, qz);

    const float u1 = angs[gid * 2 + 0];
    const float u2 = angs[gid * 2 + 1];
    const float m  = ray_mask[gid] ? 1.0f : 0.0f;

    sample_and_store(u1, u2, m, s == 0,
                     rr, nx, ny, nz, vx, vy, vz,
                     tx, ty, tz, qx, qy, qz,
                     outL, gid * 3, outB, gid * 9);
}

extern "C" void kernel_launch(void* const* d_in, const int* in_sizes, int n_in,
                              void* d_out, int out_size, void* d_ws, size_t ws_size,
                              hipStream_t stream) {
    // setup_inputs order: viewdir, normal, r1, r2(unused), ray_mask, angs
    const float* viewdir  = (const float*)d_in[0];
    const float* normal   = (const float*)d_in[1];
    const float* r1       = (const float*)d_in[2];
    const int*   ray_mask = (const int*)d_in[4];
    const float* angs     = (const float*)d_in[5];

    const int  B     = in_sizes[0] / 3;
    const int  S     = in_sizes[4] / B;       // ray_mask is (B, S)
    const long total = (long)B * S;

    float* outL = (float*)d_out;
    float* outB = outL + total * 3;           // outputs concatenated: L then basis_out

    if (S == 64 && (total % 256) == 0) {
        const int blocks = (int)(total / 256);
        lobe_s64_kernel<<<blocks, 256, 0, stream>>>(viewdir, normal, r1, ray_mask,
                                                    angs, outL, outB);
    } else {
        const int blocks = (int)((total + 255) / 256);
        lobe_generic_kernel<<<blocks, 256, 0, stream>>>(viewdir, normal, r1, ray_mask,
                                                        angs, outL, outB, S, total);
    }
}